// BackwardStep_49117245997650
// MI455X (gfx1250) — compile-verified
//
#include <hip/hip_runtime.h>
#include <hip/hip_bf16.h>

typedef __attribute__((ext_vector_type(2))) float v2f;
typedef __attribute__((ext_vector_type(8))) float v8f;
typedef __attribute__((ext_vector_type(4))) int   v4i;

#define D      512
#define MI     512
#define ME     128
#define MTOT   640
#define NBATCH 64
#define NITERS 100
#define KCH    32

// ---------------------------------------------------------------------------
// CDNA5 async global->LDS copy (ASYNCcnt-tracked, bypasses VGPRs), with a
// synchronous fallback so the TU always compiles.
// Probe-learned signature: (v4i AS1* global_src, v4i AS3* lds_dst, imm, imm).
// ---------------------------------------------------------------------------
#if defined(__has_builtin)
#  if __has_builtin(__builtin_amdgcn_global_load_async_to_lds_b128)
#    define HAVE_ASYNC_B128 1
#  endif
#endif

#define AS1 __attribute__((address_space(1)))
#define AS3 __attribute__((address_space(3)))

__device__ __forceinline__ void async_copy16(const float* __restrict__ g, float* l)
{
#if defined(HAVE_ASYNC_B128)
    AS1 v4i* gp = (AS1 v4i*)(uintptr_t)g;               // global AS ptr (non-const)
    AS3 v4i* lp = (AS3 v4i*)(uint32_t)(uintptr_t)l;     // low 32b = LDS offset
    __builtin_amdgcn_global_load_async_to_lds_b128(gp, lp, 0, 0);
#else
    *(float4*)l = *(const float4*)g;                    // sync fallback
#endif
}

__device__ __forceinline__ void wait_async_all()
{
#if defined(HAVE_ASYNC_B128)
#  if __has_builtin(__builtin_amdgcn_s_wait_asynccnt)
    __builtin_amdgcn_s_wait_asynccnt(0);
#  else
    asm volatile("s_wait_asynccnt 0x0" ::: "memory");
#  endif
#endif
}

// ---------------------------------------------------------------------------
// Kernel 1 (and 4): C[n] = alpha*(S1^T S1 + S2^T S2) [+ Q] [+ I]
// WG = 256 threads = 8 waves; WG tile 128(M) x 64(N); wave tile 32x32
// (2x2 grid of 16x16 f32 WMMA accumulators). K staged in 32-deep chunks,
// double-buffered in LDS via async global->LDS loads: chunk c+1 streams in
// while WMMAs consume chunk c; one barrier per chunk.
// LDS layout [k][col] with +8 float row pad: 16B-aligned b128 async targets,
// and fragment reads are bank-conflict-free (2*rowstride = 16 mod 64 banks).
// ---------------------------------------------------------------------------
__global__ __launch_bounds__(256)
void syrk_wmma_kernel(const float* __restrict__ s1, int m1,
                      const float* __restrict__ s2, int m2,
                      const float* __restrict__ qmat, int add_diag, float alpha,
                      float* __restrict__ dst)
{
    __shared__ __attribute__((aligned(16))) float LA[2][KCH][136];
    __shared__ __attribute__((aligned(16))) float LB[2][KCH][72];

    const int n    = blockIdx.z;
    const int i0   = blockIdx.y * 128;   // output row block
    const int j0   = blockIdx.x * 64;    // output col block
    const int tid  = threadIdx.x;
    const int lane = tid & 31;
    const int wave = tid >> 5;           // 0..7
    const int rm   = (wave & 3) * 32;    // wave row offset inside WG tile
    const int cn   = (wave >> 2) * 32;   // wave col offset inside WG tile

    const float* s1n = s1 + (size_t)n * (size_t)m1 * D;
    const float* s2n = (m2 > 0) ? (s2 + (size_t)n * (size_t)m2 * D) : nullptr;

    v8f acc00 = {}; v8f acc01 = {}; v8f acc10 = {}; v8f acc11 = {};

    const int mlane = lane & 15;  // M (A-frag) / N (B-frag) index of this lane
    const int khalf = lane >> 4;  // K-half select (K {0,1} vs K {2,3})

    const int ktot = m1 + m2;
    const int nch  = ktot / KCH;  // 512 and 128 are both multiples of 32

    // Issue one k-chunk's async loads into LDS buffer b.
    auto issue = [&](int chunk, int b) {
        const int k0 = chunk * KCH;
        const float* src; int krow0;
        if (k0 < m1) { src = s1n; krow0 = k0; }
        else         { src = s2n; krow0 = k0 - m1; }
        // A^T panel: LA[b][kk][c] = src[krow0+kk][i0+c], 16B per lane
        for (int idx = tid; idx < (KCH * 128) / 4; idx += 256) {
            int c4 = (idx & 31) * 4, kk = idx >> 5;
            async_copy16(src + (size_t)(krow0 + kk) * D + i0 + c4, &LA[b][kk][c4]);
        }
        // B panel: LB[b][kk][c] = src[krow0+kk][j0+c]
        for (int idx = tid; idx < (KCH * 64) / 4; idx += 256) {
            int c4 = (idx & 15) * 4, kk = idx >> 4;
            async_copy16(src + (size_t)(krow0 + kk) * D + j0 + c4, &LB[b][kk][c4]);
        }
    };

    issue(0, 0);  // prologue

    for (int c = 0; c < nch; ++c) {
        const int cur = c & 1;
        wait_async_all();          // this wave's chunk-c loads have landed
        __syncthreads();           // everyone's have; prev buffer free to reuse
        if (c + 1 < nch) issue(c + 1, cur ^ 1);

#pragma unroll
        for (int kk0 = 0; kk0 < KCH; kk0 += 4) {
            const int ka = kk0 + 2 * khalf;
            v2f a0, a1, b0, b1;
            a0.x = LA[cur][ka][rm + mlane];      a0.y = LA[cur][ka + 1][rm + mlane];
            a1.x = LA[cur][ka][rm + 16 + mlane]; a1.y = LA[cur][ka + 1][rm + 16 + mlane];
            b0.x = LB[cur][ka][cn + mlane];      b0.y = LB[cur][ka + 1][cn + mlane];
            b1.x = LB[cur][ka][cn + 16 + mlane]; b1.y = LB[cur][ka + 1][cn + 16 + mlane];
            acc00 = __builtin_amdgcn_wmma_f32_16x16x4_f32(false, a0, false, b0,
                        (short)0, acc00, false, false);
            acc01 = __builtin_amdgcn_wmma_f32_16x16x4_f32(false, a0, false, b1,
                        (short)0, acc01, false, false);
            acc10 = __builtin_amdgcn_wmma_f32_16x16x4_f32(false, a1, false, b0,
                        (short)0, acc10, false, false);
            acc11 = __builtin_amdgcn_wmma_f32_16x16x4_f32(false, a1, false, b1,
                        (short)0, acc11, false, false);
        }
    }

    // Epilogue. C/D layout: VGPR r holds M=r (lanes 0-15) and M=8+r (16-31),
    // N = lane%16.
    const float* qn = qmat ? (qmat + (size_t)n * D * D) : nullptr;
    float* dn = dst + (size_t)n * D * D;
    const int lrow = khalf * 8;
    const int lcol = mlane;

#define SYRK_EPILOGUE(ACC, TR, TC)                                         \
    do {                                                                   \
        _Pragma("unroll")                                                  \
        for (int r = 0; r < 8; ++r) {                                      \
            int gr = i0 + rm + (TR) + lrow + r;                            \
            int gc = j0 + cn + (TC) + lcol;                                \
            float v = alpha * (ACC)[r];                                    \
            if (qn) v += qn[(size_t)gr * D + gc];                          \
            if (add_diag && gr == gc) v += 1.0f;                           \
            dn[(size_t)gr * D + gc] = v;                                   \
        }                                                                  \
    } while (0)

    SYRK_EPILOGUE(acc00, 0, 0);
    SYRK_EPILOGUE(acc01, 0, 16);
    SYRK_EPILOGUE(acc10, 16, 0);
    SYRK_EPILOGUE(acc11, 16, 16);
#undef SYRK_EPILOGUE
}

// ---------------------------------------------------------------------------
// Kernel 2: in-place unblocked Cholesky, one workgroup per batch element.
// Column j cached in LDS; trailing update parallel over rows. M stays in L2.
// ---------------------------------------------------------------------------
__global__ __launch_bounds__(256)
void chol_kernel(float* __restrict__ Mb)
{
    __shared__ float colj[D];
    __shared__ float invd;
    float* M = Mb + (size_t)blockIdx.x * D * D;
    const int tid = threadIdx.x;

    for (int j = 0; j < D; ++j) {
        if (tid == 0) {
            float djj = sqrtf(M[(size_t)j * D + j]);
            M[(size_t)j * D + j] = djj;
            invd = 1.0f / djj;
        }
        __syncthreads();
        for (int i = j + 1 + tid; i < D; i += 256) {
            float v = M[(size_t)i * D + j] * invd;
            M[(size_t)i * D + j] = v;
            colj[i] = v;
        }
        __syncthreads();
        for (int i = j + 1 + tid; i < D; i += 256) {
            float lij = colj[i];
            float* row = M + (size_t)i * D;
            for (int k = j + 1; k <= i; ++k) row[k] -= lij * colj[k];
        }
        __syncthreads();
    }
}

// ---------------------------------------------------------------------------
// Kernel 3: W = L^{-1} (lower triangular inverse). Columns are independent:
// one thread per column, zero barriers. Upper triangle zeroed so the WMMA
// W^T W pass can run over the full K range.
// ---------------------------------------------------------------------------
__global__ __launch_bounds__(512)
void trinv_kernel(const float* __restrict__ Lb, float* __restrict__ Wb)
{
    const int n = blockIdx.x, j = threadIdx.x;
    const float* L = Lb + (size_t)n * D * D;
    float* W = Wb + (size_t)n * D * D;

    for (int i = 0; i < j; ++i) W[(size_t)i * D + j] = 0.0f;
    W[(size_t)j * D + j] = 1.0f / L[(size_t)j * D + j];
    for (int i = j + 1; i < D; ++i) {
        const float* Li = L + (size_t)i * D;
        float s = 0.0f;
        for (int k = j; k < i; ++k) s += Li[k] * W[(size_t)k * D + j];
        W[(size_t)i * D + j] = -s / Li[i];
    }
}

// ---------------------------------------------------------------------------
// Kernel 5: 100 ADMM iterations, one workgroup (512 threads) per batch.
// All GEMVs column-coalesced (Minv is symmetric); A*xs uses wave-parallel
// rows with __shfl_xor (warpSize==32 on gfx1250) reductions.
// ---------------------------------------------------------------------------
__global__ __launch_bounds__(512)
void admm_kernel(const float* __restrict__ q,  const float* __restrict__ Ai,
                 const float* __restrict__ bi, const float* __restrict__ Ae,
                 const float* __restrict__ be, const float* __restrict__ x,
                 const float* __restrict__ Minv, float* __restrict__ out)
{
    __shared__ float q2S[D], tS[MTOT], rhsS[D], xsS[D];
    __shared__ float zS[MTOT], wS[MTOT], uS[MTOT];

    const int n = blockIdx.x, tid = threadIdx.x;
    const float* Ain = Ai + (size_t)n * MI * D;
    const float* Aen = Ae + (size_t)n * ME * D;
    const float* Mn  = Minv + (size_t)n * D * D;

    q2S[tid] = q[(size_t)n * D + tid] - x[(size_t)n * D + tid];  // q2 = q - alpha*x
    for (int m = tid; m < MTOT; m += D) {
        zS[m] = 0.0f; wS[m] = 0.0f;
        uS[m] = (m < MI) ? bi[(size_t)n * MI + m] : be[(size_t)n * ME + (m - MI)];
    }
    __syncthreads();

    const int wid = tid >> 5, lane = tid & 31;
    const float NEG_INF = -__builtin_inff();

    for (int it = 0; it < NITERS; ++it) {
        for (int m = tid; m < MTOT; m += D) tS[m] = zS[m] - wS[m];
        __syncthreads();

        // rhs = -q2 + rho * A^T (z - w)   (rho = 1); thread per column, coalesced
        float s = 0.0f;
        for (int m = 0; m < MI; ++m) {
            __builtin_prefetch(&Ain[(size_t)((m + 16) & (MI - 1)) * D + tid], 0, 0);
            s += Ain[(size_t)m * D + tid] * tS[m];
        }
        for (int m = 0; m < ME; ++m) s += Aen[(size_t)m * D + tid] * tS[MI + m];
        rhsS[tid] = s - q2S[tid];
        __syncthreads();

        // xs = Minv * rhs; Minv symmetric -> column access, coalesced
        float xv = 0.0f;
        for (int jj = 0; jj < D; ++jj) xv += Mn[(size_t)jj * D + tid] * rhsS[jj];
        xsS[tid] = xv;
        __syncthreads();

        // Ax = A * xs; wave-parallel over rows, lane-parallel dot + reduction
        for (int m = wid; m < MTOT; m += 16) {
            const float* row = (m < MI) ? (Ain + (size_t)m * D)
                                        : (Aen + (size_t)(m - MI) * D);
            float p = 0.0f;
            for (int i = lane; i < D; i += 32) p += row[i] * xsS[i];
#pragma unroll
            for (int off = 16; off > 0; off >>= 1) p += __shfl_xor(p, off, 32);
            if (lane == 0) {
                float axw = p + wS[m];                 // Ax + w
                float u   = uS[m];
                float lo  = (m < MI) ? NEG_INF : u;    // l = -inf (ineq) / b_eq
                float zn  = fminf(fmaxf(axw, lo), u);  // clip
                wS[m] = axw - zn;                      // w + Ax - z_new
                zS[m] = zn;
            }
        }
        __syncthreads();
    }

    out[(size_t)n * D + tid] = xsS[tid];
}

// ---------------------------------------------------------------------------
extern "C" void kernel_launch(void* const* d_in, const int* in_sizes, int n_in,
                              void* d_out, int out_size, void* d_ws, size_t ws_size,
                              hipStream_t stream)
{
    const float* Q  = (const float*)d_in[0];
    const float* q  = (const float*)d_in[1];
    const float* Ai = (const float*)d_in[2];
    const float* bi = (const float*)d_in[3];
    const float* Ae = (const float*)d_in[4];
    const float* be = (const float*)d_in[5];
    const float* x  = (const float*)d_in[6];
    float* out = (float*)d_out;

    float* buf0 = (float*)d_ws;                        // M -> L -> Minv (64 MB)
    float* buf1 = buf0 + (size_t)NBATCH * D * D;       // W = L^{-1}    (64 MB)

    dim3 gg(D / 64, D / 128, NBATCH);

    // 1) M = Q + I + A_ineq^T A_ineq + A_eq^T A_eq   (WMMA f32 SYRK)
    syrk_wmma_kernel<<<gg, 256, 0, stream>>>(Ai, MI, Ae, ME, Q, 1, 1.0f, buf0);
    // 2) L = chol(M), in place
    chol_kernel<<<NBATCH, 256, 0, stream>>>(buf0);
    // 3) W = L^{-1}
    trinv_kernel<<<NBATCH, D, 0, stream>>>(buf0, buf1);
    // 4) Minv = W^T W   (same WMMA SYRK, single source, no Q, no diag)
    syrk_wmma_kernel<<<gg, 256, 0, stream>>>(buf1, D, nullptr, 0, nullptr, 0, 1.0f, buf0);
    // 5) 100 ADMM iterations, output last xs
    admm_kernel<<<NBATCH, D, 0, stream>>>(q, Ai, bi, Ae, be, x, buf0, out);
}